// SP_31980326486805
// MI455X (gfx1250) — compile-verified
//
#include <hip/hip_runtime.h>
#include <hip/hip_bf16.h>

// Problem constants (match reference)
#define B_    8
#define C_    128
#define N_    65536
#define S_    32

#define TPB   256
#define VEC   4
#define CHUNK (TPB * VEC)        // 1024 n-values per block
#define NCHUNK (N_ / CHUNK)      // 64 chunks per batch

typedef float v4f __attribute__((ext_vector_type(4)));
typedef int   v4i __attribute__((ext_vector_type(4)));

// ---- gfx1250 LDS float-max atomic (no return) ----------------------------
// No "memory" clobber: the only compiler-visible consumer of acc[] sits
// behind an explicit s_wait_dscnt + __syncthreads(), and keeping the asm
// side-effect-free w.r.t. compiler memory lets global loads pipeline ahead
// of the DS atomics.
__device__ __forceinline__ void lds_max_f32(float* p, float v) {
#if defined(__HIP_DEVICE_COMPILE__) && defined(__gfx1250__)
    // Generic pointer to __shared__ -> low 32 bits are the LDS byte offset.
    unsigned off = (unsigned)(unsigned long long)p;
    asm volatile("ds_max_num_f32 %0, %1" :: "v"(off), "v"(v));
#else
    *p = fmaxf(*p, v);   // host-parse fallback; never executed on device
#endif
}

__global__ __launch_bounds__(TPB) void init_out_kernel(unsigned int* __restrict__ out) {
    int i = blockIdx.x * TPB + threadIdx.x;
    if (i < B_ * C_ * S_) out[i] = 0u;   // 0u == bits of 0.0f (clamp-at-zero init)
}

__global__ __launch_bounds__(TPB) void slice_maxpool_kernel(
        const float* __restrict__ in,        // [B, C, N]
        const int*   __restrict__ slice_idx, // [B, N]
        unsigned int* __restrict__ out)      // [B, C, S] as u32 bits of f32 >= 0
{
    __shared__ float acc[C_ * S_];       // 16 KB accumulator tile, acc[c*32 + s]
    __shared__ int   idx_stage[CHUNK];   // 4 KB index staging (async-to-LDS target)

    const int tid   = threadIdx.x;
    const int b     = blockIdx.x / NCHUNK;
    const int chunk = blockIdx.x % NCHUNK;
    const int n0    = chunk * CHUNK;

    // Zero the per-block accumulator (>=0 invariant enables u32 global atomics).
    for (int i = tid; i < C_ * S_; i += TPB) acc[i] = 0.0f;
    __syncthreads();

    // ---- Stage this chunk's slice indices: gfx1250 async global->LDS -----
    const int* gidx = slice_idx + (size_t)b * N_ + n0 + tid * VEC;
    int s0, s1, s2, s3;
#if defined(__HIP_DEVICE_COMPILE__) && defined(__gfx1250__)
    {
        unsigned lds_off = (unsigned)(unsigned long long)(&idx_stage[tid * VEC]);
        asm volatile("global_load_async_to_lds_b128 %0, %1, off"
                     :: "v"(lds_off), "v"(gidx) : "memory");
        asm volatile("s_wait_asynccnt 0" ::: "memory");
        v4i si = *reinterpret_cast<v4i*>(&idx_stage[tid * VEC]);
        s0 = si.x; s1 = si.y; s2 = si.z; s3 = si.w;
    }
#else
    {
        v4i si = *reinterpret_cast<const v4i*>(gidx);
        s0 = si.x; s1 = si.y; s2 = si.z; s3 = si.w;
    }
#endif

    // ---- Stream all 128 channels for this n-window -----------------------
    // Per iteration: one coalesced non-temporal global_load_b128 (input is a
    // 256 MB single-touch stream > 192 MB L2, so don't cache it) followed by
    // 4x ds_max_num_f32. Lanes in a wave share c and spread s=idx across 32
    // distinct LDS banks; only equal-idx lanes serialize (same address).
    const float* base = in + ((size_t)b * C_) * N_ + n0 + tid * VEC;
    #pragma unroll 8
    for (int c = 0; c < C_; ++c) {
        v4f x = __builtin_nontemporal_load(
                    reinterpret_cast<const v4f*>(base + (size_t)c * N_));
        float* row = &acc[c * S_];
        lds_max_f32(&row[s0], x.x);
        lds_max_f32(&row[s1], x.y);
        lds_max_f32(&row[s2], x.z);
        lds_max_f32(&row[s3], x.w);
    }

#if defined(__HIP_DEVICE_COMPILE__) && defined(__gfx1250__)
    // Inline-asm DS ops are not tracked by the compiler: drain DScnt before
    // signaling the barrier so all lanes' LDS maxes are visible.
    asm volatile("s_wait_dscnt 0" ::: "memory");
#endif
    __syncthreads();

    // ---- Merge block-partial tile into global output ---------------------
    // acc values are >= 0, so u32 compare == f32 compare on the bit patterns.
    unsigned int* out_b = out + (size_t)b * (C_ * S_);
    for (int i = tid; i < C_ * S_; i += TPB) {
        atomicMax(&out_b[i], __float_as_uint(acc[i]));
    }
}

extern "C" void kernel_launch(void* const* d_in, const int* in_sizes, int n_in,
                              void* d_out, int out_size, void* d_ws, size_t ws_size,
                              hipStream_t stream) {
    const float* in   = (const float*)d_in[0];   // [8,128,65536,1] f32
    const int*   sidx = (const int*)d_in[1];     // [8,65536] i32
    unsigned int* out = (unsigned int*)d_out;    // [8,128,32,1] f32 (as u32 bits)

    init_out_kernel<<<(B_ * C_ * S_ + TPB - 1) / TPB, TPB, 0, stream>>>(out);
    slice_maxpool_kernel<<<B_ * NCHUNK, TPB, 0, stream>>>(in, sidx, out);
}